// ts_cov_14362370638089
// MI455X (gfx1250) — compile-verified
//
#include <hip/hip_runtime.h>

#define B_ 256
#define T_ 512
#define F_ 32
#define W_ 5
#define S_ (T_ - W_ + 1)       // 508
#define C_ (F_ * (F_ - 1) / 2) // 496
#define STILE 32
#define CPAD 497               // 497 mod 64 odd -> conflict-free strided LDS reads
#define NTHREADS 256
#define NWAVES (NTHREADS / 32)
#define SPW (STILE / NWAVES)   // 4 windows per wave

typedef float v2f __attribute__((ext_vector_type(2)));
typedef float v8f __attribute__((ext_vector_type(8)));

// pair index for (i<j) matching np.triu_indices(F, k=1) order
__device__ __forceinline__ int cidx(int i, int j) {
    return i * (2 * F_ - i - 1) / 2 + (j - i - 1);
}

__global__ __launch_bounds__(NTHREADS) void ts_cov_kernel(const float* __restrict__ x,
                                                          float* __restrict__ out) {
    __shared__ float lds[STILE * CPAD];

    const int tid  = threadIdx.x;
    const int lane = tid & 31;
    const int wave = tid >> 5;
    const int half = lane >> 4;   // 0: lanes 0-15 (K=0/1), 1: lanes 16-31 (K=2/3)
    const int l15  = lane & 15;

    const int b  = blockIdx.y;
    const int s0 = blockIdx.x * STILE;

    const float* xb = x + (size_t)b * T_ * F_;

    for (int k = 0; k < SPW; ++k) {
        const int sl = wave * SPW + k;       // wave-uniform
        const int s  = s0 + sl;              // wave-uniform
        if (s < S_) {                        // uniform guard: EXEC all-1s inside
            const float* base = xb + (size_t)s * F_;

            // WMMA f32 16x16x4 panels, K = time-in-window.
            // VGPR0 holds K=0 (lanes 0-15) / K=2 (lanes 16-31); VGPR1 holds K=1 / K=3.
            const int c0 = l15;       // feature cols 0..15
            const int c1 = l15 + 16;  // feature cols 16..31
            v2f p0, p1, r0, r1;
            p0.x = base[(half ? 2 : 0) * F_ + c0];
            p0.y = base[(half ? 3 : 1) * F_ + c0];
            p1.x = base[(half ? 2 : 0) * F_ + c1];
            p1.y = base[(half ? 3 : 1) * F_ + c1];
            // remainder pass: K=0 row = t=4, all other K rows zero.
            // Load unconditionally (t=4 always in range), select with cndmask —
            // avoids divergent exec-masked load regions.
            const float t4a = base[4 * F_ + c0];
            const float t4b = base[4 * F_ + c1];
            r0.x = half ? 0.0f : t4a;
            r0.y = 0.0f;
            r1.x = half ? 0.0f : t4b;
            r1.y = 0.0f;

            // per-column window mean: lane L owns feature column L
            float msum = 0.0f;
#pragma unroll
            for (int t = 0; t < W_; ++t) msum += base[t * F_ + lane];
            const float meanv = msum * (1.0f / W_);

            // Gram tiles: G = X_win^T * X_win  (panel serves as both A and B)
            v8f g00 = {}, g01 = {}, g11 = {};
            g00 = __builtin_amdgcn_wmma_f32_16x16x4_f32(false, p0, false, p0, (short)0, g00, false, false);
            g00 = __builtin_amdgcn_wmma_f32_16x16x4_f32(false, r0, false, r0, (short)0, g00, false, false);
            g01 = __builtin_amdgcn_wmma_f32_16x16x4_f32(false, p0, false, p1, (short)0, g01, false, false);
            g01 = __builtin_amdgcn_wmma_f32_16x16x4_f32(false, r0, false, r1, (short)0, g01, false, false);
            g11 = __builtin_amdgcn_wmma_f32_16x16x4_f32(false, p1, false, p1, (short)0, g11, false, false);
            g11 = __builtin_amdgcn_wmma_f32_16x16x4_f32(false, r1, false, r1, (short)0, g11, false, false);

            // column means needed by this lane's C/D elements (j = N + 16*jb)
            const float mj_lo = __shfl(meanv, l15);
            const float mj_hi = __shfl(meanv, l15 + 16);

            float* row = lds + sl * CPAD;
#pragma unroll
            for (int r = 0; r < 8; ++r) {
                // C/D layout: VGPR r -> M = r (lanes 0-15) or r+8 (lanes 16-31), N = l15
                const int ilo = r + half * 8;
                const int ihi = ilo + 16;
                const float mi_lo = __shfl(meanv, ilo);
                const float mi_hi = __shfl(meanv, ihi);
                // tile (0,0): i = ilo, j = l15 — strict upper triangle only
                if (ilo < l15)
                    row[cidx(ilo, l15)] = g00[r] * (1.0f / W_) - mi_lo * mj_lo;
                // tile (0,1): i = ilo, j = 16 + l15 — always i < j
                row[cidx(ilo, l15 + 16)] = g01[r] * (1.0f / W_) - mi_lo * mj_hi;
                // tile (1,1): i = ihi, j = 16 + l15 — strict upper triangle only
                if (ihi < l15 + 16)
                    row[cidx(ihi, l15 + 16)] = g11[r] * (1.0f / W_) - mi_hi * mj_hi;
            }
        }
    }
    __syncthreads();

    // Coalesced write-out: out[b][c][s], lanes sweep s (32 floats = 128B per row chunk)
    float* ob = out + (size_t)b * C_ * S_;
    for (int idx = tid; idx < C_ * STILE; idx += NTHREADS) {
        const int c  = idx >> 5;
        const int sl = idx & 31;
        const int s  = s0 + sl;
        if (s < S_) ob[(size_t)c * S_ + s] = lds[sl * CPAD + c];
    }
}

extern "C" void kernel_launch(void* const* d_in, const int* in_sizes, int n_in,
                              void* d_out, int out_size, void* d_ws, size_t ws_size,
                              hipStream_t stream) {
    (void)in_sizes; (void)n_in; (void)out_size; (void)d_ws; (void)ws_size;
    const float* x = (const float*)d_in[0];
    float* out = (float*)d_out;
    dim3 grid((S_ + STILE - 1) / STILE, B_);
    dim3 block(NTHREADS);
    ts_cov_kernel<<<grid, block, 0, stream>>>(x, out);
}